// EdgeCorrGNN_41412074668562
// MI455X (gfx1250) — compile-verified
//
#include <hip/hip_runtime.h>

typedef __attribute__((ext_vector_type(2))) float v2f;
typedef __attribute__((ext_vector_type(8))) float v8f;

#define NNODES 100000
#define NEDGES 1600000

// ---------------- degree / norm prep ----------------

__global__ void k_init_deg(float* __restrict__ deg, int n) {
  int i = blockIdx.x * blockDim.x + threadIdx.x;
  if (i < n) deg[i] = 1.0f;  // self-loop weight = 1
}

__global__ void k_deg_edges(const long long* __restrict__ dst,
                            const float* __restrict__ ew,
                            float* __restrict__ deg, int e) {
  int i = blockIdx.x * blockDim.x + threadIdx.x;
  if (i < e) unsafeAtomicAdd(&deg[(int)dst[i]], ew[i]);
}

__global__ void k_rsqrt_inplace(float* __restrict__ deg, int n) {
  int i = blockIdx.x * blockDim.x + threadIdx.x;
  if (i < n) {
    float d = deg[i];
    deg[i] = (d > 0.0f) ? rsqrtf(d) : 0.0f;
  }
}

// ---------------- fp32 WMMA GEMM: H = (relu?)A[N,K] x W[K,dout] ----------------
// One wave per 16x16 output tile. N = 6250*16 exactly, K % 4 == 0, dout % 16 == 0.
// A frag (16x4 f32, 2 VGPR): lanes 0-15 -> K={k,k+1}, lanes 16-31 -> K={k+2,k+3}.
// B frag (4x16 f32, 2 VGPR): mirrored layout with N across lanes.
// C/D (16x16 f32, 8 VGPR): VGPR m: lanes 0-15 -> M=m, lanes 16-31 -> M=m+8.

__global__ void k_gemm_wmma(const float* __restrict__ A,
                            const float* __restrict__ W,
                            float* __restrict__ H,
                            int K, int dout, int relu) {
  const int lane = threadIdx.x & 31;
  const int hl   = lane >> 4;        // 0: K pair {0,1}; 1: K pair {2,3}
  const int l    = lane & 15;
  const int row  = blockIdx.x * 16 + l;   // A row for this lane
  const int col  = blockIdx.y * 16 + l;   // B/D column for this lane

  v8f c = {};
  for (int kk = 0; kk < K; kk += 4) {
    const int ka = kk + 2 * hl;
    v2f a, b;
    a.x = A[(long long)row * K + ka];
    a.y = A[(long long)row * K + ka + 1];
    if (relu) {
      a.x = fmaxf(a.x, 0.0f);
      a.y = fmaxf(a.y, 0.0f);
    }
    b.x = W[(long long)ka * dout + col];
    b.y = W[(long long)(ka + 1) * dout + col];
    // emits v_wmma_f32_16x16x4_f32
    c = __builtin_amdgcn_wmma_f32_16x16x4_f32(
        /*neg_a=*/false, a, /*neg_b=*/false, b,
        /*c_mod=*/(short)0, c, /*reuse_a=*/false, /*reuse_b=*/false);
  }

  const int rbase = blockIdx.x * 16 + hl * 8;
#pragma unroll
  for (int m = 0; m < 8; ++m)
    H[(long long)(rbase + m) * dout + col] = c[m];
}

// ---------------- aggregation ----------------
// out[i,:] = h[i,:]*dis[i]^2 + b   (self loop + bias), then edges atomically add.

__global__ void k_agg_init(const float* __restrict__ h,
                           const float* __restrict__ dis,
                           const float* __restrict__ bias,
                           float* __restrict__ out, int d, int total) {
  int idx = blockIdx.x * blockDim.x + threadIdx.x;
  if (idx < total) {
    int i = idx / d;
    int j = idx - i * d;
    float s = dis[i];
    out[idx] = h[idx] * s * s + bias[j];
  }
}

// One wave per edge; lanes stride the feature dimension. Atomics land in L2
// (out fits in the 192MB global L2) as global_atomic_add_f32.
__global__ void k_agg_edges(const long long* __restrict__ src,
                            const long long* __restrict__ dst,
                            const float* __restrict__ ew,
                            const float* __restrict__ dis,
                            const float* __restrict__ h,
                            float* __restrict__ out, int d, int e) {
  const int widx = (blockIdx.x * blockDim.x + threadIdx.x) >> 5;
  const int lane = threadIdx.x & 31;
  if (widx >= e) return;
  const int s = (int)src[widx];
  const int t = (int)dst[widx];
  const float norm = dis[s] * ew[widx] * dis[t];
  const float* __restrict__ hs = h + (long long)s * d;
  float* __restrict__ ot = out + (long long)t * d;
  for (int j = lane; j < d; j += 32)
    unsafeAtomicAdd(&ot[j], hs[j] * norm);
}

// ---------------- tiny output head: [N,256] x [256,3] + b ----------------

__global__ void k_head(const float* __restrict__ h,
                       const float* __restrict__ Wout,
                       const float* __restrict__ bout,
                       float* __restrict__ out, int n) {
  int i = blockIdx.x * blockDim.x + threadIdx.x;
  if (i >= n) return;
  float a0 = bout[0], a1 = bout[1], a2 = bout[2];
  const float* __restrict__ hi = h + (long long)i * 256;
#pragma unroll 4
  for (int k = 0; k < 256; ++k) {
    float v = hi[k];
    a0 += v * Wout[k * 3 + 0];
    a1 += v * Wout[k * 3 + 1];
    a2 += v * Wout[k * 3 + 2];
  }
  out[i * 3 + 0] = a0;
  out[i * 3 + 1] = a1;
  out[i * 3 + 2] = a2;
}

// ---------------- launcher ----------------

extern "C" void kernel_launch(void* const* d_in, const int* in_sizes, int n_in,
                              void* d_out, int out_size, void* d_ws, size_t ws_size,
                              hipStream_t stream) {
  const float*      x    = (const float*)d_in[0];
  const long long*  ei   = (const long long*)d_in[1];   // [2, E] int64
  const float*      ew   = (const float*)d_in[2];
  const float*      W1   = (const float*)d_in[3];
  const float*      b1   = (const float*)d_in[4];
  const float*      W2   = (const float*)d_in[5];
  const float*      b2   = (const float*)d_in[6];
  const float*      W3   = (const float*)d_in[7];
  const float*      b3   = (const float*)d_in[8];
  const float*      W4   = (const float*)d_in[9];
  const float*      b4   = (const float*)d_in[10];
  const float*      Wout = (const float*)d_in[11];
  const float*      bout = (const float*)d_in[12];
  float* out = (float*)d_out;

  const long long* src = ei;            // row 0
  const long long* dst = ei + NEDGES;   // row 1

  // Workspace: two N x 256 ping-pong buffers + dis[N]
  char* ws = (char*)d_ws;
  float* bufH = (float*)ws;                                   // GEMM output
  float* bufG = (float*)(ws + (size_t)NNODES * 256 * 4);      // aggregation output
  float* dis  = (float*)(ws + (size_t)NNODES * 256 * 4 * 2);  // deg -> rsqrt(deg)

  const int TPB = 256;
  const int nBlkNodes = (NNODES + TPB - 1) / TPB;
  const int nBlkEdges = (NEDGES + TPB - 1) / TPB;
  const int nBlkEdgeWaves = (NEDGES + 7) / 8;  // 8 waves (edges) per 256-thread block

  // --- normalization coefficients ---
  k_init_deg<<<nBlkNodes, TPB, 0, stream>>>(dis, NNODES);
  k_deg_edges<<<nBlkEdges, TPB, 0, stream>>>(dst, ew, dis, NEDGES);
  k_rsqrt_inplace<<<nBlkNodes, TPB, 0, stream>>>(dis, NNODES);

  // --- layer 1: 128 -> 64 (no relu on A) ---
  {
    const int K = 128, D = 64;
    dim3 g(NNODES / 16, D / 16);
    k_gemm_wmma<<<g, 32, 0, stream>>>(x, W1, bufH, K, D, 0);
    int total = NNODES * D;
    k_agg_init<<<(total + TPB - 1) / TPB, TPB, 0, stream>>>(bufH, dis, b1, bufG, D, total);
    k_agg_edges<<<nBlkEdgeWaves, TPB, 0, stream>>>(src, dst, ew, dis, bufH, bufG, D, NEDGES);
  }
  // --- layer 2: 64 -> 128 (relu fused into A load) ---
  {
    const int K = 64, D = 128;
    dim3 g(NNODES / 16, D / 16);
    k_gemm_wmma<<<g, 32, 0, stream>>>(bufG, W2, bufH, K, D, 1);
    int total = NNODES * D;
    k_agg_init<<<(total + TPB - 1) / TPB, TPB, 0, stream>>>(bufH, dis, b2, bufG, D, total);
    k_agg_edges<<<nBlkEdgeWaves, TPB, 0, stream>>>(src, dst, ew, dis, bufH, bufG, D, NEDGES);
  }
  // --- layer 3: 128 -> 256 ---
  {
    const int K = 128, D = 256;
    dim3 g(NNODES / 16, D / 16);
    k_gemm_wmma<<<g, 32, 0, stream>>>(bufG, W3, bufH, K, D, 1);
    int total = NNODES * D;
    k_agg_init<<<(total + TPB - 1) / TPB, TPB, 0, stream>>>(bufH, dis, b3, bufG, D, total);
    k_agg_edges<<<nBlkEdgeWaves, TPB, 0, stream>>>(src, dst, ew, dis, bufH, bufG, D, NEDGES);
  }
  // --- layer 4: 256 -> 256 (no relu after aggregation) ---
  {
    const int K = 256, D = 256;
    dim3 g(NNODES / 16, D / 16);
    k_gemm_wmma<<<g, 32, 0, stream>>>(bufG, W4, bufH, K, D, 1);
    int total = NNODES * D;
    k_agg_init<<<(total + TPB - 1) / TPB, TPB, 0, stream>>>(bufH, dis, b4, bufG, D, total);
    k_agg_edges<<<nBlkEdgeWaves, TPB, 0, stream>>>(src, dst, ew, dis, bufH, bufG, D, NEDGES);
  }

  // --- output head ---
  k_head<<<(NNODES + 127) / 128, 128, 0, stream>>>(bufG, Wout, bout, out, NNODES);
}